// CompGCN_40810779247267
// MI455X (gfx1250) — compile-verified
//
#include <hip/hip_runtime.h>
#include <hip/hip_bf16.h>

#define DIM 96
#define NTILES (DIM / 16)   // 6
#define KPAIRS (DIM / 2)    // 48

typedef __attribute__((ext_vector_type(2))) float v2f;
typedef __attribute__((ext_vector_type(4))) float v4f;
typedef __attribute__((ext_vector_type(8))) float v8f;

// ---------------------------------------------------------------------------
// Zero the workspace accumulators (deg_in, deg_out, sum_in, sum_out).
// ---------------------------------------------------------------------------
__global__ void zero_kernel(float* __restrict__ p, long long n) {
    long long i = (long long)blockIdx.x * blockDim.x + threadIdx.x;
    long long stride = (long long)gridDim.x * blockDim.x;
    for (; i < n; i += stride) p[i] = 0.0f;
}

// ---------------------------------------------------------------------------
// Scatter: blockDim = (96, 4). Lane x = feature dim, y = edge within block.
// src/dst are uniform across each 96-thread row; edge_embs reads coalesced.
// Two GLOBAL_ATOMIC_ADD_F32 per (edge, dim); dim-0 threads bump degrees.
// ---------------------------------------------------------------------------
__global__ void __launch_bounds__(384)
scatter_kernel(const float* __restrict__ edge_embs,
               const int* __restrict__ src,
               const int* __restrict__ dst,
               float* __restrict__ sum_in,
               float* __restrict__ sum_out,
               float* __restrict__ deg_in,
               float* __restrict__ deg_out,
               int n_edges) {
    const int e = blockIdx.x * 4 + threadIdx.y;
    if (e >= n_edges) return;
    const int d = threadIdx.x;
    const float v = edge_embs[(long long)e * DIM + d];
    const int s = src[e];
    const int t = dst[e];
    atomicAdd(&sum_in[(long long)t * DIM + d], v);
    atomicAdd(&sum_out[(long long)s * DIM + d], v);
    if (d == 0) {
        atomicAdd(&deg_in[t], 1.0f);
        atomicAdd(&deg_out[s], 1.0f);
    }
}

// ---------------------------------------------------------------------------
// Node GEMM: h[v] = (deg_in[v]*x[v] - sum_in[v]) @ W_O
//                 + (deg_out[v]*x[v] - sum_out[v]) @ W_I + (b_O + b_I)
// One wave32 per 16-node tile; f32 WMMA 16x16x4, K=96 in 24 steps, 6 N-tiles.
// Weights staged in LDS in *operand order*: for column n and K-pair p,
// sW2[(n*KPAIRS + p)*4 + {0,1,2,3}] = {W_O[2p][n], W_O[2p+1][n],
//                                      W_I[2p][n], W_I[2p+1][n]},
// so one aligned ds_load_b128 returns both B-fragments directly in
// consecutive VGPR pairs (no repacking movs). Bias folded into acc init.
// ---------------------------------------------------------------------------
__global__ void __launch_bounds__(128)
node_gemm_kernel(const float* __restrict__ node_embs,
                 const float* __restrict__ sum_in,
                 const float* __restrict__ sum_out,
                 const float* __restrict__ deg_in,
                 const float* __restrict__ deg_out,
                 const float* __restrict__ W_O, const float* __restrict__ b_O,
                 const float* __restrict__ W_I, const float* __restrict__ b_I,
                 float* __restrict__ h, int n_nodes, int n_tiles) {
    __shared__ float sW2[DIM * KPAIRS * 4];  // 72 KB, operand-ordered
    __shared__ float sBias[DIM];
    for (int i = threadIdx.x; i < DIM * DIM; i += blockDim.x) {
        const int k = i / DIM;
        const int n = i - k * DIM;
        const int p = k >> 1;
        const int r = k & 1;
        sW2[(n * KPAIRS + p) * 4 + r]     = W_O[i];
        sW2[(n * KPAIRS + p) * 4 + 2 + r] = W_I[i];
    }
    for (int i = threadIdx.x; i < DIM; i += blockDim.x) sBias[i] = b_O[i] + b_I[i];
    __syncthreads();

    const int wave = threadIdx.x >> 5;
    const int lane = threadIdx.x & 31;
    const int tile = blockIdx.x * (blockDim.x >> 5) + wave;
    if (tile >= n_tiles) return;

    const int m    = lane & 15;   // A row within tile / B-N within N-tile
    const int half = lane >> 4;   // K sub-offset selector (0 -> K+0/1, 1 -> K+2/3)
    const int row  = tile * 16 + m;
    const int r    = (row < n_nodes) ? row : (n_nodes - 1);

    const float di = deg_in[r];
    const float dq = deg_out[r];
    const float* __restrict__ xrow  = node_embs + (long long)r * DIM;
    const float* __restrict__ sirow = sum_in   + (long long)r * DIM;
    const float* __restrict__ sorow = sum_out  + (long long)r * DIM;

    // Accumulators initialized with the bias (broadcast along M).
    v8f acc[NTILES];
#pragma unroll
    for (int t = 0; t < NTILES; ++t) {
        const float bv = sBias[t * 16 + m];
#pragma unroll
        for (int j = 0; j < 8; ++j) acc[t][j] = bv;
    }

    for (int k0 = 0; k0 < DIM; k0 += 4) {
        const int kc = k0 + 2 * half;       // this lane's K pair start
        const int p  = kc >> 1;             // K-pair index
        const v2f xe = *(const v2f*)(xrow + kc);
        const v2f si = *(const v2f*)(sirow + kc);
        const v2f so = *(const v2f*)(sorow + kc);
        v2f aO, aI;
        aO.x = di * xe.x - si.x;  aO.y = di * xe.y - si.y;
        aI.x = dq * xe.x - so.x;  aI.y = dq * xe.y - so.y;
#pragma unroll
        for (int t = 0; t < NTILES; ++t) {
            const int n = t * 16 + m;
            const v4f q = *(const v4f*)&sW2[(n * KPAIRS + p) * 4];  // 16B aligned
            v2f bO, bI;
            bO.x = q.x; bO.y = q.y;   // W_O[kc][n], W_O[kc+1][n]  (regs 0:1)
            bI.x = q.z; bI.y = q.w;   // W_I[kc][n], W_I[kc+1][n]  (regs 2:3)
            acc[t] = __builtin_amdgcn_wmma_f32_16x16x4_f32(false, aO, false, bO,
                                                           (short)0, acc[t], false, false);
            acc[t] = __builtin_amdgcn_wmma_f32_16x16x4_f32(false, aI, false, bI,
                                                           (short)0, acc[t], false, false);
        }
    }

    // D layout: reg j -> M = tile*16 + j + 8*half, N = t*16 + m
    if (tile * 16 + 16 <= n_nodes) {  // full tile: unguarded store burst
#pragma unroll
        for (int t = 0; t < NTILES; ++t) {
#pragma unroll
            for (int j = 0; j < 8; ++j) {
                const int M = tile * 16 + j + 8 * half;
                h[(long long)M * DIM + t * 16 + m] = acc[t][j];
            }
        }
    } else {  // ragged tail tile
#pragma unroll
        for (int t = 0; t < NTILES; ++t) {
#pragma unroll
            for (int j = 0; j < 8; ++j) {
                const int M = tile * 16 + j + 8 * half;
                if (M < n_nodes) h[(long long)M * DIM + t * 16 + m] = acc[t][j];
            }
        }
    }
}

// ---------------------------------------------------------------------------
// Edge GEMM: he = edge_embs @ W_rel + b_rel. Same wave-tile scheme; W_rel is
// staged transposed in LDS so each B-fragment is one ds_load_b64 landing in
// an operand-ready VGPR pair.
// ---------------------------------------------------------------------------
__global__ void __launch_bounds__(128)
edge_gemm_kernel(const float* __restrict__ edge_embs,
                 const float* __restrict__ W_rel, const float* __restrict__ b_rel,
                 float* __restrict__ he, int n_edges, int n_tiles) {
    __shared__ float sWt[DIM * DIM];  // transposed: sWt[n*DIM + k] = W[k][n]
    __shared__ float sBias[DIM];
    for (int i = threadIdx.x; i < DIM * DIM; i += blockDim.x) {
        const int k = i / DIM;
        const int n = i - k * DIM;
        sWt[n * DIM + k] = W_rel[i];
    }
    for (int i = threadIdx.x; i < DIM; i += blockDim.x) sBias[i] = b_rel[i];
    __syncthreads();

    const int wave = threadIdx.x >> 5;
    const int lane = threadIdx.x & 31;
    const int tile = blockIdx.x * (blockDim.x >> 5) + wave;
    if (tile >= n_tiles) return;

    const int m    = lane & 15;
    const int half = lane >> 4;
    const int row  = tile * 16 + m;
    const int r    = (row < n_edges) ? row : (n_edges - 1);
    const float* __restrict__ arow = edge_embs + (long long)r * DIM;

    v8f acc[NTILES];
#pragma unroll
    for (int t = 0; t < NTILES; ++t) {
        const float bv = sBias[t * 16 + m];
#pragma unroll
        for (int j = 0; j < 8; ++j) acc[t][j] = bv;
    }

    for (int k0 = 0; k0 < DIM; k0 += 4) {
        const int kc = k0 + 2 * half;
        const v2f a = *(const v2f*)(arow + kc);
#pragma unroll
        for (int t = 0; t < NTILES; ++t) {
            const v2f b = *(const v2f*)&sWt[(t * 16 + m) * DIM + kc];  // 8B aligned
            acc[t] = __builtin_amdgcn_wmma_f32_16x16x4_f32(false, a, false, b,
                                                           (short)0, acc[t], false, false);
        }
    }

    if (tile * 16 + 16 <= n_edges) {
#pragma unroll
        for (int t = 0; t < NTILES; ++t) {
#pragma unroll
            for (int j = 0; j < 8; ++j) {
                const int M = tile * 16 + j + 8 * half;
                he[(long long)M * DIM + t * 16 + m] = acc[t][j];
            }
        }
    } else {
#pragma unroll
        for (int t = 0; t < NTILES; ++t) {
#pragma unroll
            for (int j = 0; j < 8; ++j) {
                const int M = tile * 16 + j + 8 * half;
                if (M < n_edges) he[(long long)M * DIM + t * 16 + m] = acc[t][j];
            }
        }
    }
}

// ---------------------------------------------------------------------------
// Launch
// ---------------------------------------------------------------------------
extern "C" void kernel_launch(void* const* d_in, const int* in_sizes, int n_in,
                              void* d_out, int out_size, void* d_ws, size_t ws_size,
                              hipStream_t stream) {
    const float* node_embs = (const float*)d_in[0];
    const float* edge_embs = (const float*)d_in[1];
    const int*   src       = (const int*)d_in[2];
    const int*   dst       = (const int*)d_in[3];
    const float* W_O       = (const float*)d_in[4];
    const float* b_O       = (const float*)d_in[5];
    const float* W_I       = (const float*)d_in[6];
    const float* b_I       = (const float*)d_in[7];
    const float* W_rel     = (const float*)d_in[8];
    const float* b_rel     = (const float*)d_in[9];

    const int n_nodes = in_sizes[0] / DIM;
    const int n_edges = in_sizes[2];

    // Workspace layout (floats): deg_in | deg_out | sum_in | sum_out
    float* deg_in  = (float*)d_ws;
    float* deg_out = deg_in + n_nodes;
    float* sum_in  = deg_out + n_nodes;
    float* sum_out = sum_in + (long long)n_nodes * DIM;
    const long long ws_floats = 2LL * n_nodes + 2LL * n_nodes * DIM;

    float* h  = (float*)d_out;                 // n_nodes * DIM
    float* he = h + (long long)n_nodes * DIM;  // n_edges * DIM

    // 1) zero accumulators
    zero_kernel<<<4096, 256, 0, stream>>>((float*)d_ws, ws_floats);

    // 2) scatter segment-sums + degrees
    {
        dim3 blk(DIM, 4, 1);
        int blocks = (n_edges + 3) / 4;
        scatter_kernel<<<blocks, blk, 0, stream>>>(edge_embs, src, dst, sum_in,
                                                   sum_out, deg_in, deg_out, n_edges);
    }

    // 3) node transform (two fused WMMA GEMMs)
    {
        int tiles = (n_nodes + 15) / 16;
        int blocks = (tiles + 3) / 4;  // 4 wave32 per block
        node_gemm_kernel<<<blocks, 128, 0, stream>>>(node_embs, sum_in, sum_out,
                                                     deg_in, deg_out, W_O, b_O,
                                                     W_I, b_I, h, n_nodes, tiles);
    }

    // 4) edge transform (streaming WMMA GEMM)
    {
        int tiles = (n_edges + 15) / 16;
        int blocks = (tiles + 3) / 4;
        edge_gemm_kernel<<<blocks, 128, 0, stream>>>(edge_embs, W_rel, b_rel, he,
                                                     n_edges, tiles);
    }
}